// LIFLegacy_23218593202606
// MI455X (gfx1250) — compile-verified
//
#include <hip/hip_runtime.h>
#include <cstdint>

// LIF neuron scan: out[t] = spike of mem_t where
//   mem = TAU*mem + x[t];  spike = (mem > THRESH);  mem = spike ? 0 : mem;
// Shape [T=8, B=64, C=128, H=32, W=32] fp32 -> purely HBM-bound streaming scan.

typedef float v4f __attribute__((ext_vector_type(4)));
typedef int   v4i __attribute__((vector_size(16)));          // matches builtin pointee
typedef __attribute__((address_space(1))) v4i glob_v4i;      // global (AS1)
typedef __attribute__((address_space(3))) v4i lds_v4i;       // LDS (AS3)

#define T_STEPS 8
#define TAU     0.25f
#define THRESH  1.0f
#define BLOCK   256

__device__ __forceinline__ float lif_step(float& m, float xv) {
    m = __builtin_fmaf(TAU, m, xv);      // mem = tau*mem + x_t
    const bool fire = m > THRESH;        // heaviside(mem - v_th)
    m = fire ? 0.0f : m;                 // hard reset on spike
    return fire ? 1.0f : 0.0f;
}

__global__ __launch_bounds__(BLOCK) void lif_scan_kernel(
    const float* __restrict__ x,   // [T, Nspat]
    float* __restrict__ out,       // [T, Nspat]
    long long nvec4,               // Nspat / 4
    long long nspat)               // spatial element count
{
    // 8 time-slices x 256 threads x 16B = 32 KB staged per block.
    __shared__ v4f tile[T_STEPS * BLOCK];

    const long long i = (long long)blockIdx.x * BLOCK + threadIdx.x;  // vec4 id
    if (i >= nvec4) return;
    const int tid = threadIdx.x;

    // Stage all 8 strided time-slices for this thread's 4 lanes into LDS via
    // gfx1250 async global->LDS copies (ASYNCcnt-tracked, no VGPR landing pads).
#pragma unroll
    for (int t = 0; t < T_STEPS; ++t) {
        const v4f* gsrc = (const v4f*)(x + (long long)t * nspat) + i;
        v4f* lds = &tile[t * BLOCK + tid];
#if __has_builtin(__builtin_amdgcn_global_load_async_to_lds_b128)
        __builtin_amdgcn_global_load_async_to_lds_b128(
            (glob_v4i*)gsrc, (lds_v4i*)lds, /*imm offset=*/0, /*cpol=*/0);
#else
        const unsigned lds_addr = (unsigned)(unsigned long long)lds;
        asm volatile("global_load_async_to_lds_b128 %0, %1, off"
                     :
                     : "v"(lds_addr), "v"((unsigned long long)gsrc)
                     : "memory");
#endif
    }

    // Drain the async pipe before touching the staged tile.
#if __has_builtin(__builtin_amdgcn_s_wait_asynccnt)
    __builtin_amdgcn_s_wait_asynccnt(0);
    asm volatile("" ::: "memory");   // compiler barrier: no hoisting of LDS reads
#else
    asm volatile("s_wait_asynccnt 0" ::: "memory");
#endif

    // Membrane state in scalar VGPRs (vector elements can't bind to float&).
    float m0 = 0.0f, m1 = 0.0f, m2 = 0.0f, m3 = 0.0f;
#pragma unroll
    for (int t = 0; t < T_STEPS; ++t) {
        const v4f xv = tile[t * BLOCK + tid];   // ds_load_b128
        v4f sp;
        sp.x = lif_step(m0, xv.x);
        sp.y = lif_step(m1, xv.y);
        sp.z = lif_step(m2, xv.z);
        sp.w = lif_step(m3, xv.w);
        // Spikes are write-once streams: non-temporal 128-bit stores.
        v4f* gout = (v4f*)(out + (long long)t * nspat) + i;
        __builtin_nontemporal_store(sp, gout);
    }
}

extern "C" void kernel_launch(void* const* d_in, const int* in_sizes, int n_in,
                              void* d_out, int out_size, void* d_ws, size_t ws_size,
                              hipStream_t stream) {
    (void)n_in; (void)out_size; (void)d_ws; (void)ws_size;
    const float* x = (const float*)d_in[0];
    float* out = (float*)d_out;

    const long long total = (long long)in_sizes[0];     // T * Nspat = 67,108,864
    const long long nspat = total / T_STEPS;            // 8,388,608
    const long long nvec4 = nspat / 4;                  // 2,097,152

    const long long nblocks = (nvec4 + BLOCK - 1) / BLOCK;   // 8192
    lif_scan_kernel<<<dim3((unsigned)nblocks), dim3(BLOCK), 0, stream>>>(
        x, out, nvec4, nspat);
}